// WormVAE_18700287607571
// MI455X (gfx1250) — compile-verified
//
#include <hip/hip_runtime.h>

// ---------------------------------------------------------------------------
// WormVAE calcium pathway, MI455X (gfx1250, wave32).
//
//   x[t]   = a[t] / (tau+eps)
//   c[t]   = c[t-1]*exp(-dt/tau) + x[t]          (length-8192 scan per row)
//   out[t] = scale*( dt*c[t] + init*exp(-t*dt/tau) ) + shift
//
// Memory-bound (160MB @ 23.3TB/s ~ 7us). One block per (b,n) row.
// Tile-local scans done as triangular 16x16 matmuls on the WMMA pipe
// (V_WMMA_F32_16X16X4_F32 chained over K), carries combined hierarchically
// with v_readlane broadcasts (no LDS round-trips on the carry chain).
// Input/output staged through LDS via gfx1250 async global<->LDS DMA.
// ---------------------------------------------------------------------------

typedef float v2f __attribute__((ext_vector_type(2)));
typedef float v8f __attribute__((ext_vector_type(8)));

#define DT_STEP    0.25f
#define EPS_F      1e-8f
#define T_LEN      8192
#define T_INIT_N   10
#define TPB        256
#define NWAVES     (TPB / 32)              // 8 waves (wave32)
#define WAVE_ELEMS (T_LEN / NWAVES)        // 1024 timesteps per wave
#define TILES      (WAVE_ELEMS / 256)      // 4 tiles of 256 per wave
#define PAD_ROW    17                      // 16 floats + 1 pad  -> conflict-free
#define TILE_PAD   (16 * PAD_ROW)          // 272 floats per padded tile
#define WAVE_PAD   ((WAVE_ELEMS / 16) * PAD_ROW)  // 1088
#define LDS_FLOATS ((T_LEN / 16) * PAD_ROW)       // 8704 floats = 34816 B

// ---- gfx1250 async global<->LDS path (guarded; falls back to plain ops) ----
#if __has_builtin(__builtin_amdgcn_global_load_async_to_lds_b32) && \
    __has_builtin(__builtin_amdgcn_global_store_async_from_lds_b32)
#define HAVE_ASYNC 1
#else
#define HAVE_ASYNC 0
#endif

typedef __attribute__((address_space(1))) int as1_int;
typedef __attribute__((address_space(3))) int as3_int;

// Builtin params are (addrspace(1) int*, addrspace(3) int*, imm offset, imm cpol).
__device__ __forceinline__ as1_int* as_global_int(const void* p) {
  return (as1_int*)(int*)const_cast<void*>(p);
}
__device__ __forceinline__ as3_int* as_shared_int(void* p) {
  return (as3_int*)(int*)p;
}

__device__ __forceinline__ void wait_asynccnt0() {
#if __has_builtin(__builtin_amdgcn_s_wait_asynccnt)
  __builtin_amdgcn_s_wait_asynccnt(0);
#else
  asm volatile("s_wait_asynccnt 0x0" ::: "memory");
#endif
  asm volatile("" ::: "memory");  // compiler barrier: LDS reads stay below
}

__device__ __forceinline__ void wait_dscnt0() {
  asm volatile("" ::: "memory");  // compiler barrier: LDS writes stay above
#if __has_builtin(__builtin_amdgcn_s_wait_dscnt)
  __builtin_amdgcn_s_wait_dscnt(0);
#else
  asm volatile("s_wait_dscnt 0x0" ::: "memory");
#endif
  asm volatile("" ::: "memory");
}

// Wave-uniform broadcast from a compile-time lane: v_readlane_b32 (no LDS trip).
__device__ __forceinline__ float readlane_f(float v, int srcLane) {
  return __int_as_float(__builtin_amdgcn_readlane(__float_as_int(v), srcLane));
}

__global__ __launch_bounds__(TPB) void wormvae_calcium_scan(
    const float* __restrict__ act, const float* __restrict__ fluor,
    const float* __restrict__ scale, const float* __restrict__ shift,
    const float* __restrict__ tau_p, float* __restrict__ out, int N)
{
  __shared__ float lds_buf[LDS_FLOATS];   // padded x, later reused for output
  __shared__ float waveSums[NWAVES];
  __shared__ float initSh;

  const int row  = blockIdx.x;            // row = b*N + n
  const int nidx = row % N;
  const int tid  = threadIdx.x;
  const int lane = tid & 31;
  const int w    = tid >> 5;
  const int half = lane >> 4;             // 0: lanes 0-15, 1: lanes 16-31
  const int nseg = lane & 15;             // A-row m / B,D-column n for WMMA

  const float tau   = tau_p[0];
  const float alpha = DT_STEP / tau;              // dt / tau
  const float invte = 1.0f / (tau + EPS_F);       // x = a * invte (folded into L)
  const float sc    = scale[nidx];
  const float sh    = shift[nidx];

  const size_t rbase = (size_t)row * (size_t)T_LEN;
  const float* aRow  = act + rbase;
  float*       oRow  = out + rbase;

  const int wbase = w * WAVE_ELEMS;       // this wave's timestep base
  const int pbase = w * WAVE_PAD;         // its padded LDS base

  // ---- stage this wave's 1024 inputs into padded LDS (async DMA path) ----
#if HAVE_ASYNC
#pragma unroll
  for (int it = 0; it < WAVE_ELEMS / 32; ++it) {
    const int g = wbase + it * 32 + lane;                 // contiguous 128B/instr
    const int p = pbase + it * 34 + lane + half;          // +1 pad per 16 floats
    __builtin_amdgcn_global_load_async_to_lds_b32(
        as_global_int(aRow + g), as_shared_int(&lds_buf[p]), 0, 0);
  }
#else
#pragma unroll
  for (int it = 0; it < WAVE_ELEMS / 32; ++it) {
    const int g = wbase + it * 32 + lane;
    const int p = pbase + it * 34 + lane + half;
    lds_buf[p] = aRow[g];
  }
#endif

  // ---- init_calcium = (mean(fluor[:10]) - shift)/(scale+eps), one scalar ----
  if (tid == 0) {
    const float* fRow = fluor + rbase;
    float s = 0.f;
    for (int t = 0; t < T_INIT_N; ++t) s += fRow[t];
    initSh = (s * (1.0f / (float)T_INIT_N) - sh) / (sc + EPS_F);
  }

#if HAVE_ASYNC
  wait_asynccnt0();
#endif

  // ---- constant triangular operand L (A-matrix), inv_tau folded in --------
  // A 16x4 layout: lanes 0-15 M=lane (K=0,1 in v0,v1); lanes 16-31 (K=2,3).
  v2f A[4];
#pragma unroll
  for (int k = 0; k < 4; ++k) {
    const int m  = nseg;
    const int c0 = 4 * k + 2 * half;
    A[k].x = (m >= c0)     ? __expf(-alpha * (float)(m - c0))     * invte : 0.f;
    A[k].y = (m >= c0 + 1) ? __expf(-alpha * (float)(m - c0 - 1)) * invte : 0.f;
  }

  // decay-power helpers for carry injection: element (m,n) gets carry*d^(16n+m+1)
  float dpow[8];
#pragma unroll
  for (int r = 0; r < 8; ++r)
    dpow[r] = __expf(-alpha * (float)(r + 8 * half + 1));   // decay^(m+1)
  const float d16   = __expf(-16.0f   * alpha);
  const float dseg  = __expf(-16.0f   * alpha * (float)nseg);  // decay^(16n)
  const float d256  = __expf(-256.0f  * alpha);
  const float d1024 = __expf(-1024.0f * alpha);

  // ---- per-tile scans via WMMA, chained carries within the wave -----------
  v8f   acc[TILES];
  float s_chain = 0.f;                     // scan state entering current tile
#pragma unroll
  for (int tt = 0; tt < TILES; ++tt) {
    const int tb = pbase + tt * TILE_PAD + PAD_ROW * nseg;

    // batch the 4 B operands (conflict-free ds reads), then chain the WMMAs
    v2f b[4];
#pragma unroll
    for (int k = 0; k < 4; ++k) {
      const int cr = 4 * k + 2 * half;     // K = v + 2*(l>>4), N = l&15
      b[k].x = lds_buf[tb + cr + 0];
      b[k].y = lds_buf[tb + cr + 1];
    }
    v8f y = {0.f, 0.f, 0.f, 0.f, 0.f, 0.f, 0.f, 0.f};
#pragma unroll
    for (int k = 0; k < 4; ++k)
      y = __builtin_amdgcn_wmma_f32_16x16x4_f32(false, A[k], false, b[k],
                                                (short)0, y, false, false);

    // zero-state segment ends e[j] = Yloc[15][j] live in y[7] on lanes 16+j.
    // Broadcast all 16 via v_readlane (independent, no LDS), then Horner:
    //   carry into segment n: s_n = e[n-1] + d16*s_(n-1)
    float e_arr[16];
    const float e7 = y[7];
#pragma unroll
    for (int j = 0; j < 16; ++j) e_arr[j] = readlane_f(e7, 16 + j);
    float slane = 0.f;
#pragma unroll
    for (int j = 0; j < 16; ++j)
      if (j < nseg) slane = slane * d16 + e_arr[j];

    // inject segment carry + tile-entering state: += (s_n + s_in*d16^n)*d^(m+1)
    const float inj = slane + s_chain * dseg;
#pragma unroll
    for (int r = 0; r < 8; ++r) y[r] += inj * dpow[r];

    s_chain = readlane_f(y[7], 31);        // corrected tile-end value (uniform)
    acc[tt] = y;
  }

  // ---- cross-wave carries -------------------------------------------------
  if (lane == 0) waveSums[w] = s_chain;    // wave total, zero entering state
  __syncthreads();
  float E = 0.f;                           // state entering this wave
  for (int u = 0; u < w; ++u) E = E * d1024 + waveSums[u];
  const float init_c = initSh;

  // ---- finalize: add wave carry, affine output, stage to LDS, stream out --
  float d256p = 1.f;
#pragma unroll
  for (int tt = 0; tt < TILES; ++tt) {
    const int tb  = pbase + tt * TILE_PAD + PAD_ROW * nseg;
    const int tg0 = wbase + tt * 256 + 16 * nseg;
    const v8f y = acc[tt];
#pragma unroll
    for (int r = 0; r < 8; ++r) {
      const int m   = r + 8 * half;
      const float c = y[r] + E * d256p * dseg * dpow[r];
      const float tg = (float)(tg0 + m);
      const float val = sc * (DT_STEP * c + init_c * __expf(-alpha * tg)) + sh;
      lds_buf[tb + m] = val;               // padded staging (own region)
    }
    d256p *= d256;
  }

#if HAVE_ASYNC
  wait_dscnt0();                           // LDS writes visible to async DMA
#pragma unroll
  for (int it = 0; it < WAVE_ELEMS / 32; ++it) {
    const int g = wbase + it * 32 + lane;
    const int p = pbase + it * 34 + lane + half;
    __builtin_amdgcn_global_store_async_from_lds_b32(
        as_global_int(oRow + g), as_shared_int(&lds_buf[p]), 0, 0);
  }
  wait_asynccnt0();                        // all DMA retired before wave exit
#else
#pragma unroll
  for (int it = 0; it < WAVE_ELEMS / 32; ++it) {
    const int g = wbase + it * 32 + lane;
    const int p = pbase + it * 34 + lane + half;
    oRow[g] = lds_buf[p];
  }
#endif
}

extern "C" void kernel_launch(void* const* d_in, const int* in_sizes, int n_in,
                              void* d_out, int out_size, void* d_ws, size_t ws_size,
                              hipStream_t stream) {
  (void)n_in; (void)out_size; (void)d_ws; (void)ws_size;
  const float* act   = (const float*)d_in[0];  // (B,N,T) f32
  const float* fluor = (const float*)d_in[1];  // (B,N,T) f32
  const float* scale = (const float*)d_in[2];  // (N,)
  const float* shift = (const float*)d_in[3];  // (N,)
  const float* tau   = (const float*)d_in[4];  // (1,)
  float* out = (float*)d_out;

  const int N    = in_sizes[2];
  const int rows = in_sizes[0] / T_LEN;        // B*N = 2416

  wormvae_calcium_scan<<<dim3(rows), dim3(TPB), 0, stream>>>(
      act, fluor, scale, shift, tau, out, N);
}